// GATv2MiniModel_32804960207347
// MI455X (gfx1250) — compile-verified
//
#include <hip/hip_runtime.h>
#include <math.h>

// ---------------------------------------------------------------------------
// GATv2 hetero mini-model for MI455X (gfx1250), wave32.
// GEMMs use native fp32 WMMA (V_WMMA_F32_16X16X4_F32) with a 32x32 tile per
// wave (2x2 accumulators) so each wmma costs one 8B load per lane instead of
// two.  Edge attention phase is gather/scatter bandwidth bound: wave-per-edge
// kernels with float atomics.
// ---------------------------------------------------------------------------

typedef __attribute__((ext_vector_type(2))) float v2f;
typedef __attribute__((ext_vector_type(8))) float v8f;

#define N_NODES 100000   // Nu == Ni
#define N_EDGES 300000
#define DIM     128      // hidden dim D
#define HC2     256      // H*C = 2*128
#define NHEAD   2

// ---------------------------------------------------------------- utilities
__device__ __forceinline__ float waveReduceSum(float v) {
#pragma unroll
  for (int off = 16; off > 0; off >>= 1) v += __shfl_xor(v, off, 32);
  return v;
}

__device__ __forceinline__ void atomicMaxF(float* addr, float val) {
  int* ia = (int*)addr;
  int cur = *ia;
  while (__int_as_float(cur) < val) {
    int prev = atomicCAS(ia, cur, __float_as_int(val));
    if (prev == cur) break;
    cur = prev;
  }
}

// --------------------------------------------------------- fp32 WMMA GEMM
// C[M,N] = act(A[M,K] @ W[N,K]^T + bias[N]).  Row-major. M%32==0, N%32==0,
// K%4==0.  One wave -> one 32x32 output tile as 2x2 16x16 WMMA accumulators;
// K stepped by 4 through v_wmma_f32_16x16x4_f32.
//
// Lane layouts (CDNA5 ISA 7.12.2, 32-bit A 16x4 / B 4x16 / C 16x16):
//   A: lane&15 = M row; (lane>>4)*2 = first of 2 K values per lane.
//   B: lane&15 = N col; (lane>>4)*2 = first of 2 K values per lane.
//   C: lane&15 = N col; vgpr r holds row r + 8*(lane>>4).
template <int ACT>  // 0 none, 1 relu, 2 tanh
__global__ void gemm_wmma_f32(const float* __restrict__ A,
                              const float* __restrict__ W,
                              const float* __restrict__ bias,
                              float* __restrict__ C,
                              int M, int N, int K) {
  const int wave = (blockIdx.x * blockDim.x + threadIdx.x) >> 5;
  const int lane = threadIdx.x & 31;
  const int tilesN = N >> 5;
  const int totTiles = (M >> 5) * tilesN;
  if (wave >= totTiles) return;
  const int tm = wave / tilesN;
  const int tn = wave - tm * tilesN;

  const int khalf = (lane >> 4) << 1;  // 0 or 2

  const float* __restrict__ a0 =
      A + (size_t)((tm << 5) + (lane & 15)) * K + khalf;
  const float* __restrict__ a1 = a0 + (size_t)16 * K;
  const float* __restrict__ b0 =
      W + (size_t)((tn << 5) + (lane & 15)) * K + khalf;
  const float* __restrict__ b1 = b0 + (size_t)16 * K;

  v8f c00 = {}, c01 = {}, c10 = {}, c11 = {};
#pragma unroll 2
  for (int k = 0; k < K; k += 4) {
    v2f va0 = *(const v2f*)(a0 + k);
    v2f va1 = *(const v2f*)(a1 + k);
    v2f vb0 = *(const v2f*)(b0 + k);
    v2f vb1 = *(const v2f*)(b1 + k);
    c00 = __builtin_amdgcn_wmma_f32_16x16x4_f32(false, va0, false, vb0,
                                                (short)0, c00, false, false);
    c01 = __builtin_amdgcn_wmma_f32_16x16x4_f32(false, va0, false, vb1,
                                                (short)0, c01, false, false);
    c10 = __builtin_amdgcn_wmma_f32_16x16x4_f32(false, va1, false, vb0,
                                                (short)0, c10, false, false);
    c11 = __builtin_amdgcn_wmma_f32_16x16x4_f32(false, va1, false, vb1,
                                                (short)0, c11, false, false);
  }

  const int rbase = (lane >> 4) << 3;  // rows 0..7 or 8..15 of each 16-tile
  const int n0 = (tn << 5) + (lane & 15);
  const int n1 = n0 + 16;
  const float bn0 = bias ? bias[n0] : 0.0f;
  const float bn1 = bias ? bias[n1] : 0.0f;
#pragma unroll
  for (int r = 0; r < 8; ++r) {
    const int row0 = (tm << 5) + rbase + r;
    const int row1 = row0 + 16;
    float v00 = c00[r] + bn0, v01 = c01[r] + bn1;
    float v10 = c10[r] + bn0, v11 = c11[r] + bn1;
    if (ACT == 1) {
      v00 = v00 > 0.0f ? v00 : 0.0f; v01 = v01 > 0.0f ? v01 : 0.0f;
      v10 = v10 > 0.0f ? v10 : 0.0f; v11 = v11 > 0.0f ? v11 : 0.0f;
    }
    if (ACT == 2) {
      v00 = tanhf(v00); v01 = tanhf(v01);
      v10 = tanhf(v10); v11 = tanhf(v11);
    }
    C[(size_t)row0 * N + n0] = v00;
    C[(size_t)row0 * N + n1] = v01;
    C[(size_t)row1 * N + n0] = v10;
    C[(size_t)row1 * N + n1] = v11;
  }
}

// ------------------------------------------------------------ small kernels
__global__ void fill_f32(float* p, float v, int n) {
  int i = blockIdx.x * blockDim.x + threadIdx.x;
  if (i < n) p[i] = v;
}

// one wave per edge: logits[e,h] = att_h . leaky_relu(xl[s]+xr[d]+w*We, 0.2)
__global__ void edge_logits(const float* __restrict__ xl,
                            const float* __restrict__ xr,
                            const int* __restrict__ esrc,
                            const int* __restrict__ edst,
                            const float* __restrict__ ea,
                            const float* __restrict__ logw,   // scalar
                            const float* __restrict__ We,     // [256]
                            const float* __restrict__ att,    // [2,128]
                            float* __restrict__ elog,         // [E,2]
                            float* __restrict__ mmax,         // [Nd,2]
                            int E) {
  const int e = ((blockIdx.x * blockDim.x + threadIdx.x) >> 5);
  const int lane = threadIdx.x & 31;
  if (e >= E) return;
  const int s = esrc[e], d = edst[e];
  const float w = ea[e] * __expf(logw[0]);
  const float* xls = xl + (size_t)s * HC2;
  const float* xrd = xr + (size_t)d * HC2;
#pragma unroll
  for (int h = 0; h < NHEAD; ++h) {
    float lsum = 0.0f;
#pragma unroll
    for (int i = 0; i < 4; ++i) {
      const int c = lane + 32 * i;
      float v = xls[h * DIM + c] + xrd[h * DIM + c] + w * We[h * DIM + c];
      v = v > 0.0f ? v : 0.2f * v;
      lsum += v * att[h * DIM + c];
    }
    lsum = waveReduceSum(lsum);
    if (lane == 0) {
      elog[(size_t)e * NHEAD + h] = lsum;
      atomicMaxF(&mmax[(size_t)d * NHEAD + h], lsum);
    }
  }
}

// one thread per (edge, head): exp and denominator accumulate
__global__ void edge_exp(float* __restrict__ elog,
                         const int* __restrict__ edst,
                         const float* __restrict__ mmax,
                         float* __restrict__ denom, int E) {
  int t = blockIdx.x * blockDim.x + threadIdx.x;
  if (t >= E * NHEAD) return;
  const int e = t >> 1, h = t & 1;
  const int d = edst[e];
  const float ex = __expf(elog[t] - mmax[(size_t)d * NHEAD + h]);
  elog[t] = ex;
  atomicAdd(&denom[(size_t)d * NHEAD + h], ex);
}

// one wave per edge: acc[d] += alpha * xl[s]
__global__ void edge_scatter(const float* __restrict__ xl,
                             const int* __restrict__ esrc,
                             const int* __restrict__ edst,
                             const float* __restrict__ elog,
                             const float* __restrict__ denom,
                             float* __restrict__ acc, int E) {
  const int e = ((blockIdx.x * blockDim.x + threadIdx.x) >> 5);
  const int lane = threadIdx.x & 31;
  if (e >= E) return;
  const int s = esrc[e], d = edst[e];
  const float* xls = xl + (size_t)s * HC2;
  float* accd = acc + (size_t)d * HC2;
#pragma unroll
  for (int h = 0; h < NHEAD; ++h) {
    const float alpha = elog[(size_t)e * NHEAD + h] /
                        (denom[(size_t)d * NHEAD + h] + 1e-16f);
#pragma unroll
    for (int i = 0; i < 4; ++i) {
      const int c = h * DIM + lane + 32 * i;
      atomicAdd(&accd[c], xls[c] * alpha);
    }
  }
}

// rel[n,c] = 0.5*(acc[n,c] + acc[n,128+c]) + bias[c]   (head mean)
__global__ void head_mean(const float* __restrict__ acc,
                          const float* __restrict__ bias,
                          float* __restrict__ rel, int N) {
  int t = blockIdx.x * blockDim.x + threadIdx.x;
  if (t >= N * DIM) return;
  const int n = t >> 7, c = t & 127;
  rel[t] = 0.5f * (acc[(size_t)n * HC2 + c] + acc[(size_t)n * HC2 + DIM + c]) +
           bias[c];
}

// one wave per node: glog[n*R + r] = tanh_feats[n] . w2
__global__ void gate_logit(const float* __restrict__ t,
                           const float* __restrict__ w2,
                           float* __restrict__ glog, int R, int r, int N) {
  const int n = ((blockIdx.x * blockDim.x + threadIdx.x) >> 5);
  const int lane = threadIdx.x & 31;
  if (n >= N) return;
  float s = 0.0f;
#pragma unroll
  for (int i = 0; i < 4; ++i) {
    const int c = lane + 32 * i;
    s += t[(size_t)n * DIM + c] * w2[c];
  }
  s = waveReduceSum(s);
  if (lane == 0) glog[(size_t)n * R + r] = s;
}

// one block (128 threads) per node: relation softmax + residual + LN + relu
__global__ void gate_combine(const float* __restrict__ h,
                             const float* __restrict__ rel0,
                             const float* __restrict__ rel1,
                             const float* __restrict__ glog, int R,
                             const float* __restrict__ gamma,
                             const float* __restrict__ beta,
                             float* __restrict__ out) {
  const int n = blockIdx.x, c = threadIdx.x;
  float w0 = 1.0f, w1 = 0.0f;
  if (R == 2) {
    const float l0 = glog[(size_t)n * 2], l1 = glog[(size_t)n * 2 + 1];
    const float mx = fmaxf(l0, l1);
    const float e0 = __expf(l0 - mx), e1 = __expf(l1 - mx);
    const float inv = 1.0f / (e0 + e1);
    w0 = e0 * inv;
    w1 = e1 * inv;
  }
  float agg = w0 * rel0[(size_t)n * DIM + c];
  if (R == 2) agg += w1 * rel1[(size_t)n * DIM + c];
  const float z = h[(size_t)n * DIM + c] + agg;

  __shared__ float red[DIM];
  red[c] = z;
  __syncthreads();
#pragma unroll
  for (int s = 64; s > 0; s >>= 1) {
    if (c < s) red[c] += red[c + s];
    __syncthreads();
  }
  const float mu = red[0] * (1.0f / DIM);
  __syncthreads();
  const float dz = z - mu;
  red[c] = dz * dz;
  __syncthreads();
#pragma unroll
  for (int s = 64; s > 0; s >>= 1) {
    if (c < s) red[c] += red[c + s];
    __syncthreads();
  }
  const float var = red[0] * (1.0f / DIM);
  float v = dz * rsqrtf(var + 1e-5f) * gamma[c] + beta[c];
  out[(size_t)n * DIM + c] = v > 0.0f ? v : 0.0f;
}

// one block (128 threads) per node: L2 normalize
__global__ void l2norm(const float* __restrict__ y, float* __restrict__ out) {
  const int n = blockIdx.x, c = threadIdx.x;
  const float v = y[(size_t)n * DIM + c];
  __shared__ float red[DIM];
  red[c] = v * v;
  __syncthreads();
#pragma unroll
  for (int s = 64; s > 0; s >>= 1) {
    if (c < s) red[c] += red[c + s];
    __syncthreads();
  }
  const float inv = 1.0f / fmaxf(sqrtf(red[0]), 1e-12f);
  out[(size_t)n * DIM + c] = v * inv;
}

// ------------------------------------------------------------- host helpers
static void launch_gemm(int act, const float* A, const float* W,
                        const float* bias, float* C, int M, int N, int K,
                        hipStream_t s) {
  const int tiles = (M >> 5) * (N >> 5);
  const int blocks = (tiles + 3) / 4;  // 4 waves (128 threads) per block
  if (act == 0)
    gemm_wmma_f32<0><<<blocks, 128, 0, s>>>(A, W, bias, C, M, N, K);
  else if (act == 1)
    gemm_wmma_f32<1><<<blocks, 128, 0, s>>>(A, W, bias, C, M, N, K);
  else
    gemm_wmma_f32<2><<<blocks, 128, 0, s>>>(A, W, bias, C, M, N, K);
}

// Input index map: jax pytree flatten (dict keys sorted alphabetically).
//  0 ea_clicks  1 ea_rated  2 ea_rates
//  3 ei_clicks  4 ei_rated  5 ei_rates            (int32 [2,E])
//  6 final.item.W  7 final.item.b  8 final.user.W  9 final.user.b
//  layer block (31 arrays) at 10 + 31*layer:
//    +0..6   conv.item__rated_by__user : We,Wl,Wr,att,bias,bl,br
//    +7..13  conv.user__clicks__item   : We,Wl,Wr,att,bias,bl,br
//    +14..20 conv.user__rates__item    : We,Wl,Wr,att,bias,bl,br
//    +21..23 gate.item: W1,b1,w2   +24..26 gate.user: W1,b1,w2
//    +27,28  norm.item: gamma,beta +29,30 norm.user: gamma,beta
//  72 proj.item.W  73 proj.item.b  74 proj.user.W  75 proj.user.b
//  76 rel_logw.item__rated_by__user  77 ...clicks  78 ...rates
//  79 x_item [100000,64]   80 x_user [100000,128]

struct ConvScratch {
  float *xl, *xr, *acc, *elog, *mmax, *denom;
};

static void run_conv(const float* h_src, int Ns, const float* h_dst, int Nd,
                     const int* ei, const float* ea, const float* logw,
                     void* const* P, int pb, const ConvScratch& w,
                     float* rel_out, hipStream_t s) {
  const float* We   = (const float*)P[pb + 0];
  const float* Wl   = (const float*)P[pb + 1];
  const float* Wr   = (const float*)P[pb + 2];
  const float* att  = (const float*)P[pb + 3];
  const float* bias = (const float*)P[pb + 4];
  const float* bl   = (const float*)P[pb + 5];
  const float* br   = (const float*)P[pb + 6];
  const int* esrc = ei;
  const int* edst = ei + N_EDGES;

  hipMemsetAsync(w.acc, 0, (size_t)Nd * HC2 * sizeof(float), s);
  hipMemsetAsync(w.denom, 0, (size_t)Nd * NHEAD * sizeof(float), s);
  fill_f32<<<(Nd * NHEAD + 255) / 256, 256, 0, s>>>(w.mmax, -INFINITY,
                                                    Nd * NHEAD);

  launch_gemm(0, h_src, Wl, bl, w.xl, Ns, HC2, DIM, s);
  launch_gemm(0, h_dst, Wr, br, w.xr, Nd, HC2, DIM, s);

  const int eb = (N_EDGES * 32 + 255) / 256;  // wave per edge
  edge_logits<<<eb, 256, 0, s>>>(w.xl, w.xr, esrc, edst, ea, logw, We, att,
                                 w.elog, w.mmax, N_EDGES);
  edge_exp<<<(N_EDGES * NHEAD + 255) / 256, 256, 0, s>>>(w.elog, edst, w.mmax,
                                                         w.denom, N_EDGES);
  edge_scatter<<<eb, 256, 0, s>>>(w.xl, esrc, edst, w.elog, w.denom, w.acc,
                                  N_EDGES);
  head_mean<<<(Nd * DIM + 255) / 256, 256, 0, s>>>(w.acc, bias, rel_out, Nd);
}

extern "C" void kernel_launch(void* const* d_in, const int* in_sizes, int n_in,
                              void* d_out, int out_size, void* d_ws,
                              size_t ws_size, hipStream_t stream) {
  (void)in_sizes; (void)n_in; (void)out_size; (void)ws_size;

  const float* ea_rates  = (const float*)d_in[2];
  const float* ea_clicks = (const float*)d_in[0];
  const float* ea_rated  = (const float*)d_in[1];
  const int* ei_rates  = (const int*)d_in[5];
  const int* ei_clicks = (const int*)d_in[3];
  const int* ei_rated  = (const int*)d_in[4];
  const float* logw_rated  = (const float*)d_in[76];
  const float* logw_clicks = (const float*)d_in[77];
  const float* logw_rates  = (const float*)d_in[78];
  const float* x_item = (const float*)d_in[79];
  const float* x_user = (const float*)d_in[80];

  // ---- workspace carve-up (floats) ----
  float* ws = (float*)d_ws;
  size_t off = 0;
  auto carve = [&](size_t n) { float* p = ws + off; off += n; return p; };
  const size_t ND = (size_t)N_NODES * DIM;
  const size_t NH2 = (size_t)N_NODES * HC2;
  float* h_user  = carve(ND);
  float* h_item  = carve(ND);
  float* nh_user = carve(ND);
  float* nh_item = carve(ND);
  ConvScratch cs;
  cs.xl    = carve(NH2);
  cs.xr    = carve(NH2);
  cs.acc   = carve(NH2);
  cs.elog  = carve((size_t)N_EDGES * NHEAD);
  cs.mmax  = carve((size_t)N_NODES * NHEAD);
  cs.denom = carve((size_t)N_NODES * NHEAD);
  float* rel_u0 = carve(ND);
  float* rel_i0 = carve(ND);
  float* rel_i1 = carve(ND);
  float* glog   = carve((size_t)N_NODES * 2);
  float* tq   = cs.acc;  // gate tanh scratch: conv acc is free by gate time
  float* yfin = cs.xl;   // final-linear scratch: xl free after last layer

  // ---- input projection (+relu) ----
  launch_gemm(1, x_user, (const float*)d_in[74], (const float*)d_in[75],
              h_user, N_NODES, DIM, 128, stream);
  launch_gemm(1, x_item, (const float*)d_in[72], (const float*)d_in[73],
              h_item, N_NODES, DIM, 64, stream);

  const int waveN = (N_NODES * 32 + 255) / 256;  // wave per node
  for (int layer = 0; layer < 2; ++layer) {
    const int base = 10 + layer * 31;
    // conv order must match reference stacking: item gets [rates, clicks]
    run_conv(h_user, N_NODES, h_item, N_NODES, ei_rates, ea_rates, logw_rates,
             d_in, base + 14, cs, rel_i0, stream);
    run_conv(h_user, N_NODES, h_item, N_NODES, ei_clicks, ea_clicks,
             logw_clicks, d_in, base + 7, cs, rel_i1, stream);
    run_conv(h_item, N_NODES, h_user, N_NODES, ei_rated, ea_rated, logw_rated,
             d_in, base + 0, cs, rel_u0, stream);

    // gate + residual + LN + relu : item (R=2)
    const float* W1i = (const float*)d_in[base + 21];
    const float* b1i = (const float*)d_in[base + 22];
    const float* w2i = (const float*)d_in[base + 23];
    launch_gemm(2, rel_i0, W1i, b1i, tq, N_NODES, DIM, DIM, stream);
    gate_logit<<<waveN, 256, 0, stream>>>(tq, w2i, glog, 2, 0, N_NODES);
    launch_gemm(2, rel_i1, W1i, b1i, tq, N_NODES, DIM, DIM, stream);
    gate_logit<<<waveN, 256, 0, stream>>>(tq, w2i, glog, 2, 1, N_NODES);
    gate_combine<<<N_NODES, DIM, 0, stream>>>(
        h_item, rel_i0, rel_i1, glog, 2, (const float*)d_in[base + 27],
        (const float*)d_in[base + 28], nh_item);

    // gate : user (R=1)
    const float* W1u = (const float*)d_in[base + 24];
    const float* b1u = (const float*)d_in[base + 25];
    const float* w2u = (const float*)d_in[base + 26];
    launch_gemm(2, rel_u0, W1u, b1u, tq, N_NODES, DIM, DIM, stream);
    gate_logit<<<waveN, 256, 0, stream>>>(tq, w2u, glog, 1, 0, N_NODES);
    gate_combine<<<N_NODES, DIM, 0, stream>>>(
        h_user, rel_u0, nullptr, glog, 1, (const float*)d_in[base + 29],
        (const float*)d_in[base + 30], nh_user);

    float* t;
    t = h_user; h_user = nh_user; nh_user = t;
    t = h_item; h_item = nh_item; nh_item = t;
  }

  // ---- final linear + L2 normalize : outputs (user, item) concatenated ----
  float* out = (float*)d_out;
  launch_gemm(0, h_user, (const float*)d_in[8], (const float*)d_in[9], yfin,
              N_NODES, DIM, DIM, stream);
  l2norm<<<N_NODES, DIM, 0, stream>>>(yfin, out);
  launch_gemm(0, h_item, (const float*)d_in[6], (const float*)d_in[7], yfin,
              N_NODES, DIM, DIM, stream);
  l2norm<<<N_NODES, DIM, 0, stream>>>(yfin, out + ND);
}